// ContextQueryAttn_9990093930622
// MI455X (gfx1250) — compile-verified
//
#include <hip/hip_runtime.h>

typedef __attribute__((ext_vector_type(2))) float v2f;
typedef __attribute__((ext_vector_type(4))) float f4;
typedef __attribute__((ext_vector_type(8))) float v8f;

#define DIM   512
#define NCTX  1024
#define NQ    128
#define NB    64

// ws layout (float offsets)
#define WS_COLTERM 0                       // 64*128
#define WS_QSUM    (64*128)                // 64*512
#define WS_BSUM    (WS_QSUM + 64*512)      // 64*512
#define WS_PCMAX   (WS_BSUM + 64*512)      // 4*64*128
#define WS_PCSUM   (WS_PCMAX + 4*64*128)   // 4*64*128
// total = 139264 floats = 557 KB

// ---------------------------------------------------------------------------
// prep: colterm = q.wq ; qsum = sum_j q (== every A row) ; zero bsum.
// No context reads at all.  grid 64 x 256
// ---------------------------------------------------------------------------
__global__ void prep_kernel(const float* __restrict__ qry,
                            const float* __restrict__ w0, float* __restrict__ ws) {
    int b = blockIdx.x, tid = threadIdx.x;
    if (tid < NQ) {
        const f4* row = (const f4*)(qry + ((long)b * NQ + tid) * DIM);
        const f4* w4  = (const f4*)(w0 + DIM);
        float s = 0.f;
#pragma unroll 8
        for (int k = 0; k < DIM / 4; ++k) {
            f4 a = row[k], w = w4[k];
            s += a.x * w.x + a.y * w.y + a.z * w.z + a.w * w.w;
        }
        ws[WS_COLTERM + b * NQ + tid] = s;
    }
    for (int d = tid; d < DIM; d += 256) {
        float s = 0.f;
        for (int j = 0; j < NQ; ++j) s += qry[((long)b * NQ + j) * DIM + d];
        ws[WS_QSUM + b * DIM + d] = s;
    }
    for (int k = tid; k < DIM; k += 256) ws[WS_BSUM + b * DIM + k] = 0.f;
}

// ---------------------------------------------------------------------------
// 4 sim tiles (one j-tile x 4 consecutive i-tiles) via f32 WMMA.
// B'[k][j] = wm[k]*Q[j][k] + wc[k]  folds the rowterm broadcast into the GEMM:
//   C x B' = C*diag(wm)*Q^T + (C.wc)*1^T
// B operand is reused across the 4 i-tiles; 4 independent acc chains.
// Lane layout (CDNA5 16x16x4 f32): lanes 0-15 -> M=lane,K={0,1};
// lanes 16-31 -> M=lane-16,K={2,3}.  D: lane=N, VGPR r = M=r (+8 upper half).
// ---------------------------------------------------------------------------
__device__ inline void sim_group(const float* __restrict__ Crow,
                                 const float* __restrict__ Qrow,
                                 const float* __restrict__ wmS,
                                 const float* __restrict__ wcS,
                                 int half, v8f acc[4]) {
#pragma unroll 2
    for (int k0 = 0; k0 < DIM; k0 += 4) {
        int kb = k0 + 2 * half;
        v2f q  = *(const v2f*)(Qrow + kb);
        v2f wm = *(const v2f*)(wmS + kb);
        v2f wc = *(const v2f*)(wcS + kb);
        v2f bv = q * wm + wc;            // v_pk_fma_f32
#pragma unroll
        for (int t = 0; t < 4; ++t) {
            v2f a = *(const v2f*)(Crow + (long)t * 16 * DIM + kb);
            acc[t] = __builtin_amdgcn_wmma_f32_16x16x4_f32(false, a, false, bv,
                                                           (short)0, acc[t], false, false);
        }
    }
}

// ---------------------------------------------------------------------------
// colstats: per (b, j) online max / sum-exp over an i-chunk of 256 rows.
// grid (64, 4) x 256 : 8 waves = 8 j-tiles of 16, chunk = 4 groups of 4 i-tiles
// ---------------------------------------------------------------------------
__global__ void colstats_kernel(const float* __restrict__ ctx, const float* __restrict__ qry,
                                const float* __restrict__ w0, float* __restrict__ ws) {
    __shared__ float wmS[DIM], wcS[DIM];
    int b = blockIdx.x, c = blockIdx.y, tid = threadIdx.x;
    for (int k = tid; k < DIM; k += 256) {
        wmS[k] = w0[2 * DIM + k];
        wcS[k] = w0[k];
    }
    __syncthreads();
    int lane = tid & 31, wave = tid >> 5;
    int m = lane & 15, half = lane >> 4;
    int j0 = wave * 16;
    const float* Qrow = qry + ((long)b * NQ + j0 + m) * DIM;
    float ct = ws[WS_COLTERM + b * NQ + j0 + m];
    float runmax = -__builtin_inff();
    float runsum = 0.f;
    for (int g = 0; g < 4; ++g) {
        int i0 = c * 256 + g * 64;
        const float* Crow = ctx + ((long)b * NCTX + i0 + m) * DIM;
        v8f acc[4];
#pragma unroll
        for (int t = 0; t < 4; ++t) acc[t] = (v8f){0.f, 0.f, 0.f, 0.f, 0.f, 0.f, 0.f, 0.f};
        sim_group(Crow, Qrow, wmS, wcS, half, acc);
#pragma unroll
        for (int t = 0; t < 4; ++t) {
            float s[8];
            float tmax = -__builtin_inff();
#pragma unroll
            for (int r = 0; r < 8; ++r) {
                s[r] = acc[t][r] + ct;   // rowterm already folded in via wc
                tmax = fmaxf(tmax, s[r]);
            }
            tmax = fmaxf(tmax, __shfl_xor(tmax, 16, 32));
            float nm = fmaxf(runmax, tmax);
            float e = 0.f;
#pragma unroll
            for (int r = 0; r < 8; ++r) e += __expf(s[r] - nm);
            e += __shfl_xor(e, 16, 32);
            runsum = runsum * __expf(runmax - nm) + e;
            runmax = nm;
        }
    }
    if (half == 0) {
        ws[WS_PCMAX + ((c * NB + b) * NQ) + j0 + m] = runmax;
        ws[WS_PCSUM + ((c * NB + b) * NQ) + j0 + m] = runsum;
    }
}

// ---------------------------------------------------------------------------
// taccum+bsum fused: combine partial softmax stats, recompute sim tiles,
// reduce t[b,i] for this chunk entirely in LDS, then accumulate
// bb_partial[d] = sum_{i in chunk} t[i]*ctx[i,d] (chunk re-read hits L2).
// grid (64, 4) x 256
// ---------------------------------------------------------------------------
__global__ void taccum_bsum_kernel(const float* __restrict__ ctx, const float* __restrict__ qry,
                                   const float* __restrict__ w0, float* __restrict__ ws) {
    __shared__ float wmS[DIM], wcS[DIM];
    __shared__ float cmS[NQ], czS[NQ];
    __shared__ float tpart[8][256];
    __shared__ float tsum[256];
    int b = blockIdx.x, c = blockIdx.y, tid = threadIdx.x;
    for (int k = tid; k < DIM; k += 256) {
        wmS[k] = w0[2 * DIM + k];
        wcS[k] = w0[k];
    }
    if (tid < NQ) {  // combine the 4 partial (max,sum) chunks for column j=tid
        float M = -__builtin_inff(), Z = 0.f;
#pragma unroll
        for (int cc = 0; cc < 4; ++cc) {
            float mm = ws[WS_PCMAX + ((cc * NB + b) * NQ) + tid];
            float zz = ws[WS_PCSUM + ((cc * NB + b) * NQ) + tid];
            float nM = fmaxf(M, mm);
            Z = Z * __expf(M - nM) + zz * __expf(mm - nM);
            M = nM;
        }
        cmS[tid] = M;
        czS[tid] = Z;
    }
    for (int k = tid; k < 8 * 256; k += 256) ((float*)tpart)[k] = 0.f;
    __syncthreads();

    int lane = tid & 31, wave = tid >> 5;
    int m = lane & 15, half = lane >> 4;
    int j0 = wave * 16;
    const float* Qrow = qry + ((long)b * NQ + j0 + m) * DIM;
    float ct  = ws[WS_COLTERM + b * NQ + j0 + m];
    float cm  = cmS[j0 + m];
    float inv = 1.f / czS[j0 + m];
    for (int g = 0; g < 4; ++g) {
        int i0 = c * 256 + g * 64;
        const float* Crow = ctx + ((long)b * NCTX + i0 + m) * DIM;
        v8f acc[4];
#pragma unroll
        for (int t = 0; t < 4; ++t) acc[t] = (v8f){0.f, 0.f, 0.f, 0.f, 0.f, 0.f, 0.f, 0.f};
        sim_group(Crow, Qrow, wmS, wcS, half, acc);
#pragma unroll
        for (int t = 0; t < 4; ++t) {
            float pr[8];
#pragma unroll
            for (int r = 0; r < 8; ++r)
                pr[r] = __expf(acc[t][r] + ct - cm) * inv;
            // sum over the 16 j-lanes within each half-wave
#pragma unroll
            for (int r = 0; r < 8; ++r) {
                float v = pr[r];
                v += __shfl_xor(v, 1, 32);
                v += __shfl_xor(v, 2, 32);
                v += __shfl_xor(v, 4, 32);
                v += __shfl_xor(v, 8, 32);
                pr[r] = v;
            }
            if (m == 0) {
#pragma unroll
                for (int r = 0; r < 8; ++r)
                    tpart[wave][g * 64 + t * 16 + 8 * half + r] += pr[r];
            }
        }
    }
    __syncthreads();
    {   // reduce the 8 per-wave partials -> t for this chunk's 256 rows
        float tv = 0.f;
#pragma unroll
        for (int w = 0; w < 8; ++w) tv += tpart[w][tid];
        tsum[tid] = tv;
    }
    __syncthreads();
    // bb partial for this chunk (context chunk re-read served by L2)
    const float* Cb = ctx + ((long)b * NCTX + c * 256) * DIM;
#pragma unroll
    for (int dd = 0; dd < 2; ++dd) {
        int d = tid + dd * 256;
        float s = 0.f;
        for (int i = 0; i < 256; ++i) s += tsum[i] * Cb[(long)i * DIM + d];
        atomicAdd(&ws[WS_BSUM + b * DIM + d], s);
    }
}

// ---------------------------------------------------------------------------
// bcast: A[b,i,:]=qsum[b,:], B[b,i,:]=bsum[b,:].  8 rows per block, value
// loaded once, streaming non-temporal 128-bit stores.  grid 8192 x 256
// ---------------------------------------------------------------------------
__global__ void bcast_kernel(const float* __restrict__ ws, float* __restrict__ out) {
    long row0 = (long)blockIdx.x * 8;   // 8 consecutive rows share one batch b
    int b = (int)(row0 >> 10);
    int tid = threadIdx.x;
    if (tid < 128) {
        f4 v = ((const f4*)(ws + WS_QSUM + b * DIM))[tid];
        f4* dst = (f4*)out + row0 * (DIM / 4) + tid;
#pragma unroll
        for (int r = 0; r < 8; ++r)
            __builtin_nontemporal_store(v, dst + (long)r * (DIM / 4));
    } else {
        int t2 = tid - 128;
        f4 v = ((const f4*)(ws + WS_BSUM + b * DIM))[t2];
        f4* dst = (f4*)(out + (long)NB * NCTX * DIM) + row0 * (DIM / 4) + t2;
#pragma unroll
        for (int r = 0; r < 8; ++r)
            __builtin_nontemporal_store(v, dst + (long)r * (DIM / 4));
    }
}

extern "C" void kernel_launch(void* const* d_in, const int* in_sizes, int n_in,
                              void* d_out, int out_size, void* d_ws, size_t ws_size,
                              hipStream_t stream) {
    const float* ctx = (const float*)d_in[0];
    const float* qry = (const float*)d_in[1];
    const float* w0  = (const float*)d_in[2];
    float* out = (float*)d_out;
    float* ws  = (float*)d_ws;

    prep_kernel<<<NB, 256, 0, stream>>>(qry, w0, ws);
    colstats_kernel<<<dim3(NB, 4), 256, 0, stream>>>(ctx, qry, w0, ws);
    taccum_bsum_kernel<<<dim3(NB, 4), 256, 0, stream>>>(ctx, qry, w0, ws);
    bcast_kernel<<<NB * NCTX / 8, 256, 0, stream>>>(ws, out);
}